// ARPE_85040352460815
// MI455X (gfx1250) — compile-verified
//
#include <hip/hip_runtime.h>

#define FLT_MAX_C 3.402823466e+38f

typedef __attribute__((ext_vector_type(2))) float v2f;
typedef __attribute__((ext_vector_type(8))) float v8f;

// ---------------------------------------------------------------------------
// Kernel 1: brute-force KNN (32 nearest, self included), batch slice in LDS.
// Block = 128 threads (1 query each), grid = 16 batches * 16 tiles = 256.
// ---------------------------------------------------------------------------
__global__ __launch_bounds__(128) void knn_kernel(const float* __restrict__ x,
                                                  int* __restrict__ idxb) {
  __shared__ float xs0[2048], xs1[2048], xs2[2048];
  __shared__ float bd[32 * 128];
  __shared__ int   bi[32 * 128];
  const int tid  = threadIdx.x;
  const int b    = blockIdx.x >> 4;
  const int tile = blockIdx.x & 15;
  const float* xb = x + b * 2048 * 3;

  for (int i = tid; i < 2048; i += 128) {
    xs0[i] = xb[i * 3 + 0];
    xs1[i] = xb[i * 3 + 1];
    xs2[i] = xb[i * 3 + 2];
  }
  for (int k = 0; k < 32; ++k) {
    bd[k * 128 + tid] = FLT_MAX_C;
    bi[k * 128 + tid] = -1;
  }
  __syncthreads();

  const int n = tile * 128 + tid;
  const float q0 = xs0[n], q1 = xs1[n], q2 = xs2[n];
  const float sqn = fmaf(q2, q2, fmaf(q1, q1, q0 * q0));
  float worst = FLT_MAX_C;

  for (int m = 0; m < 2048; ++m) {
    const float y0 = xs0[m], y1 = xs1[m], y2 = xs2[m];
    const float dot = fmaf(q2, y2, fmaf(q1, y1, q0 * y0));
    const float sqm = fmaf(y2, y2, fmaf(y1, y1, y0 * y0));
    const float d = (sqn + sqm) - 2.0f * dot;   // same formula as reference
    if (d < worst) {                            // strict <: lower index wins ties
      int p = 31;
      while (p > 0 && bd[(p - 1) * 128 + tid] > d) {
        bd[p * 128 + tid] = bd[(p - 1) * 128 + tid];
        bi[p * 128 + tid] = bi[(p - 1) * 128 + tid];
        --p;
      }
      bd[p * 128 + tid] = d;
      bi[p * 128 + tid] = m;
      worst = bd[31 * 128 + tid];
    }
  }
  int* dst = idxb + (b * 2048 + n) * 32;
  for (int k = 0; k < 32; ++k) dst[k] = bi[k * 128 + tid];
}

// ---------------------------------------------------------------------------
// Kernel 2: per-block partial sums / sumsq of z1 = h@W1 + b1  (6 channels).
// One thread per (b,n,k) edge; deterministic LDS tree reduction.
// ---------------------------------------------------------------------------
__global__ __launch_bounds__(256) void stats1_kernel(const float* __restrict__ x,
                                                     const int* __restrict__ idxb,
                                                     const float* __restrict__ W1v,
                                                     const float* __restrict__ b1v,
                                                     float* __restrict__ part1) {
  __shared__ float red[12 * 256];
  const int tid = threadIdx.x;
  const int e = blockIdx.x * 256 + tid;       // flat (b,n,k), NK = 65536
  const int b = e >> 16;
  const int r = e & 65535;
  const int n = r >> 5;
  const int m = idxb[e];
  const float* xb = x + b * 2048 * 3;

  const float xn0 = xb[n * 3 + 0], xn1 = xb[n * 3 + 1], xn2 = xb[n * 3 + 2];
  const float xm0 = xb[m * 3 + 0], xm1 = xb[m * 3 + 1], xm2 = xb[m * 3 + 2];
  const float h[6] = {xn0, xn1, xn2, xn0 - xm0, xn1 - xm1, xn2 - xm2};

  for (int c = 0; c < 6; ++c) {
    float acc = 0.f;
    for (int j = 0; j < 6; ++j) acc = fmaf(h[j], W1v[j * 6 + c], acc);
    acc += b1v[c];
    red[c * 256 + tid]       = acc;
    red[(6 + c) * 256 + tid] = acc * acc;
  }
  __syncthreads();
  for (int off = 128; off > 0; off >>= 1) {
    if (tid < off)
      for (int c = 0; c < 12; ++c) red[c * 256 + tid] += red[c * 256 + tid + off];
    __syncthreads();
  }
  if (tid < 12) part1[blockIdx.x * 12 + tid] = red[tid * 256];
}

// Kernel 2b: reduce 4096 block partials -> 12 doubles (fixed order).
__global__ __launch_bounds__(256) void reduce1_kernel(const float* __restrict__ part1,
                                                      double* __restrict__ stats1) {
  __shared__ double acc[12 * 16];
  const int tid = threadIdx.x;
  if (tid < 192) {
    const int c = tid % 12, chunk = tid / 12;
    double s = 0.0;
    for (int i = chunk * 256; i < chunk * 256 + 256; ++i) s += (double)part1[i * 12 + c];
    acc[c * 16 + chunk] = s;
  }
  __syncthreads();
  if (tid < 12) {
    double s = 0.0;
    for (int j = 0; j < 16; ++j) s += acc[tid * 16 + j];
    stats1[tid] = s;
  }
}

// ---------------------------------------------------------------------------
// Kernel 3 (fused): recompute z1, BN1+ELU, max-pool over K=32 -> hmax[6] in LDS,
// then GEMM2 (hmax @ W2 + b2) via V_WMMA_F32_16X16X4_F32, plus BN2 partials.
// Block = 256 threads (1 point each, 8 waves), grid = 128 (exact fit, EXEC full).
// ---------------------------------------------------------------------------
__global__ __launch_bounds__(256) void fused_kernel(const float* __restrict__ x,
                                                    const int* __restrict__ idxb,
                                                    const double* __restrict__ stats1,
                                                    const float* __restrict__ W1v,
                                                    const float* __restrict__ b1v,
                                                    const float* __restrict__ g1,
                                                    const float* __restrict__ be1,
                                                    const float* __restrict__ W2v,
                                                    const float* __restrict__ b2v,
                                                    float* __restrict__ z2b,
                                                    float* __restrict__ part2) {
  __shared__ float hLds[256 * 6];
  __shared__ float contribS[32][16];
  __shared__ float contribQ[32][16];
  const int tid = threadIdx.x;
  const int row = blockIdx.x * 256 + tid;   // (b,n) flat, 32768 total
  const int b = row >> 11;
  const int n = row & 2047;
  const float* xb = x + b * 2048 * 3;

  // BN1 parameters from global stats (count = B*N*K = 1048576, biased var)
  float mean1[6], rstd1[6];
  const double inv1 = 1.0 / 1048576.0;
  for (int c = 0; c < 6; ++c) {
    const double mu = stats1[c] * inv1;
    const double va = stats1[6 + c] * inv1 - mu * mu;
    mean1[c] = (float)mu;
    rstd1[c] = rsqrtf((float)va + 1e-5f);
  }

  const float xn0 = xb[n * 3 + 0], xn1 = xb[n * 3 + 1], xn2 = xb[n * 3 + 2];
  float hmax[6];
  for (int c = 0; c < 6; ++c) hmax[c] = -FLT_MAX_C;

  for (int k = 0; k < 32; ++k) {
    const int m = idxb[row * 32 + k];
    const float xm0 = xb[m * 3 + 0], xm1 = xb[m * 3 + 1], xm2 = xb[m * 3 + 2];
    const float h[6] = {xn0, xn1, xn2, xn0 - xm0, xn1 - xm1, xn2 - xm2};
    for (int c = 0; c < 6; ++c) {
      float z = 0.f;
      for (int j = 0; j < 6; ++j) z = fmaf(h[j], W1v[j * 6 + c], z);
      z += b1v[c];
      float t = g1[c] * (z - mean1[c]);
      t = fmaf(t, rstd1[c], be1[c]);
      const float e = t > 0.f ? t : expm1f(t);   // ELU
      hmax[c] = fmaxf(hmax[c], e);
    }
  }
  for (int c = 0; c < 6; ++c) hLds[tid * 6 + c] = hmax[c];
  __syncthreads();

  // ---- WMMA GEMM: [16 rows x 6] @ [6 x 32] + b2, fp32 tensor path ----
  const int lane = tid & 31, wave = tid >> 5;
  const int halfSel = lane >> 4;   // 0: lanes 0-15 (K=0,1), 1: lanes 16-31 (K=2,3)
  const int lp = lane & 15;
  float accS[2] = {0.f, 0.f}, accQ[2] = {0.f, 0.f};

  for (int g = 0; g < 2; ++g) {
    const int group = wave * 2 + g;        // 16-row group id (0..15)
    const int arow = group * 16 + lp;
    v2f a0, a1;
    // chunk 0: k = halfSel*2 + {0,1}  (k in 0..3, always valid)
    a0.x = hLds[arow * 6 + halfSel * 2 + 0];
    a0.y = hLds[arow * 6 + halfSel * 2 + 1];
    // chunk 1: k = 4 + halfSel*2 + {0,1}  (k=4,5 valid, k=6,7 zero pad)
    a1.x = halfSel ? 0.f : hLds[arow * 6 + 4];
    a1.y = halfSel ? 0.f : hLds[arow * 6 + 5];

    for (int t = 0; t < 2; ++t) {
      const int nc = t * 16 + lp;          // output column (0..31)
      v2f b0, b1;
      b0.x = W2v[(halfSel * 2 + 0) * 32 + nc];
      b0.y = W2v[(halfSel * 2 + 1) * 32 + nc];
      b1.x = halfSel ? 0.f : W2v[4 * 32 + nc];
      b1.y = halfSel ? 0.f : W2v[5 * 32 + nc];

      const float bias = b2v[nc];
      v8f cacc;
      for (int rr = 0; rr < 8; ++rr) cacc[rr] = bias;   // C = b2 broadcast
      cacc = __builtin_amdgcn_wmma_f32_16x16x4_f32(false, a0, false, b0,
                                                   (short)0, cacc, false, false);
      cacc = __builtin_amdgcn_wmma_f32_16x16x4_f32(false, a1, false, b1,
                                                   (short)0, cacc, false, false);
      // D layout: lanes 0-15 -> M = r, lanes 16-31 -> M = r+8 (N = lane%16)
      const int mbase = blockIdx.x * 256 + group * 16 + halfSel * 8;
      float s = 0.f, q = 0.f;
      for (int rr = 0; rr < 8; ++rr) {
        const float v = cacc[rr];
        z2b[(mbase + rr) * 32 + nc] = v;
        s += v;
        q = fmaf(v, v, q);
      }
      accS[t] += s;
      accQ[t] += q;
    }
  }
  // deterministic per-block BN2 partials: unique (channel, slot) writers
  for (int t = 0; t < 2; ++t) {
    contribS[t * 16 + lp][wave * 2 + halfSel] = accS[t];
    contribQ[t * 16 + lp][wave * 2 + halfSel] = accQ[t];
  }
  __syncthreads();
  if (tid < 32) {
    float s = 0.f, q = 0.f;
    for (int j = 0; j < 16; ++j) { s += contribS[tid][j]; q += contribQ[tid][j]; }
    part2[blockIdx.x * 64 + tid]      = s;
    part2[blockIdx.x * 64 + 32 + tid] = q;
  }
}

// Kernel 3b: reduce 128 block partials -> 64 doubles (fixed order).
__global__ __launch_bounds__(64) void reduce2_kernel(const float* __restrict__ part2,
                                                     double* __restrict__ stats2) {
  const int t = threadIdx.x;   // 64 threads
  double s = 0.0;
  for (int blk = 0; blk < 128; ++blk) s += (double)part2[blk * 64 + t];
  stats2[t] = s;
}

// ---------------------------------------------------------------------------
// Kernel 4: y = ELU(BN2(z2)) elementwise over [B,N,32] = 1048576.
// ---------------------------------------------------------------------------
__global__ __launch_bounds__(256) void bn2_kernel(const float* __restrict__ z2b,
                                                  const double* __restrict__ stats2,
                                                  const float* __restrict__ g2,
                                                  const float* __restrict__ be2,
                                                  float* __restrict__ out) {
  const int gid = blockIdx.x * 256 + threadIdx.x;
  const int c = gid & 31;
  const double inv = 1.0 / 32768.0;
  const double mu = stats2[c] * inv;
  const double va = stats2[32 + c] * inv - mu * mu;
  const float rstd = rsqrtf((float)va + 1e-5f);
  const float z = z2b[gid];
  float t = g2[c] * (z - (float)mu);
  t = fmaf(t, rstd, be2[c]);
  out[gid] = t > 0.f ? t : expm1f(t);
}

// ---------------------------------------------------------------------------
extern "C" void kernel_launch(void* const* d_in, const int* in_sizes, int n_in,
                              void* d_out, int out_size, void* d_ws, size_t ws_size,
                              hipStream_t stream) {
  const float* x   = (const float*)d_in[0];
  const float* W1v = (const float*)d_in[1];
  const float* b1v = (const float*)d_in[2];
  const float* g1  = (const float*)d_in[3];
  const float* be1 = (const float*)d_in[4];
  const float* W2v = (const float*)d_in[5];
  const float* b2v = (const float*)d_in[6];
  const float* g2  = (const float*)d_in[7];
  const float* be2 = (const float*)d_in[8];
  float* out = (float*)d_out;

  // workspace layout (doubles first for alignment)
  double* stats1 = (double*)d_ws;                  // 12
  double* stats2 = stats1 + 12;                    // 64
  float*  part1  = (float*)(stats2 + 64);          // 4096 * 12
  float*  part2  = part1 + 4096 * 12;              // 128 * 64
  int*    idxb   = (int*)(part2 + 128 * 64);       // 16*2048*32
  float*  z2b    = (float*)(idxb + 16 * 2048 * 32);// 16*2048*32

  knn_kernel   <<<256,  128, 0, stream>>>(x, idxb);
  stats1_kernel<<<4096, 256, 0, stream>>>(x, idxb, W1v, b1v, part1);
  reduce1_kernel<<<1,   256, 0, stream>>>(part1, stats1);
  fused_kernel <<<128,  256, 0, stream>>>(x, idxb, stats1, W1v, b1v, g1, be1,
                                          W2v, b2v, z2b, part2);
  reduce2_kernel<<<1,    64, 0, stream>>>(part2, stats2);
  bn2_kernel   <<<4096, 256, 0, stream>>>(z2b, stats2, g2, be2, out);
}